// LatticeGNN_39926015984147
// MI455X (gfx1250) — compile-verified
//
#include <hip/hip_runtime.h>

typedef __attribute__((ext_vector_type(2))) float v2f;
typedef __attribute__((ext_vector_type(8))) float v8f;

#define HID 128

// ---------------- utility kernels ----------------
__global__ void k_zero_i32(int* __restrict__ p, int n) {
  int i = blockIdx.x * 256 + threadIdx.x;
  if (i < n) p[i] = 0;
}
__global__ void k_zero_f32(float* __restrict__ p, int n) {
  int i = blockIdx.x * 256 + threadIdx.x;
  if (i < n) p[i] = 0.0f;
}
__global__ void k_copy_i32(const int* __restrict__ a, int* __restrict__ b, int n) {
  int i = blockIdx.x * 256 + threadIdx.x;
  if (i < n) b[i] = a[i];
}
__global__ void k_wcombine(const float* __restrict__ tw, const float* __restrict__ pw,
                           float* __restrict__ wc, int n) {
  int i = blockIdx.x * 256 + threadIdx.x;
  if (i < n) wc[i] = pw[i] - tw[i];
}
__global__ void k_bcombine(const float* __restrict__ tb, const float* __restrict__ pb,
                           float* __restrict__ bc, int n) {
  int i = blockIdx.x * 256 + threadIdx.x;
  if (i < n) bc[i] = tb[i] + pb[i];
}

// ---------------- CSR build (dst shared by all 3 layers) ----------------
__global__ void k_count(const int* __restrict__ dst, int* __restrict__ cnt, int E) {
  int i = blockIdx.x * 256 + threadIdx.x;
  if (i < E) atomicAdd(&cnt[dst[i]], 1);
}

// single-workgroup tiled inclusive scan
__global__ __launch_bounds__(1024) void k_scan(const int* __restrict__ cnt,
                                               int* __restrict__ rowptr, int N) {
  __shared__ int sm[1024];
  __shared__ int carry;
  const int t = threadIdx.x;
  if (t == 0) { carry = 0; rowptr[0] = 0; }
  __syncthreads();
  for (int base = 0; base < N; base += 1024) {
    int i = base + t;
    sm[t] = (i < N) ? cnt[i] : 0;
    __syncthreads();
    for (int off = 1; off < 1024; off <<= 1) {
      int v = (t >= off) ? sm[t - off] : 0;
      __syncthreads();
      sm[t] += v;
      __syncthreads();
    }
    if (i < N) rowptr[i + 1] = carry + sm[t];
    __syncthreads();
    if (t == 0) carry += sm[1023];
    __syncthreads();
  }
}

__global__ void k_scatter(const int* __restrict__ src, const int* __restrict__ dst,
                          int* __restrict__ cursor, int* __restrict__ csr, int E) {
  int i = blockIdx.x * 256 + threadIdx.x;
  if (i < E) {
    int d = dst[i];
    int pos = atomicAdd(&cursor[d], 1);
    csr[pos] = src[i];
  }
}

// ---------------- layer 1 (K = 2, scalar is right) ----------------
__global__ void k_layer1(const float* __restrict__ coords,
                         const float* __restrict__ t1w, const float* __restrict__ t1b,
                         const float* __restrict__ p1w, const float* __restrict__ p1b,
                         float* __restrict__ TH, float* __restrict__ BT, int total) {
  int i = blockIdx.x * 256 + threadIdx.x;
  if (i >= total) return;
  int n = i >> 7, c = i & 127;
  float x0 = coords[2 * n], x1 = coords[2 * n + 1];
  float th = x0 * t1w[c] + x1 * t1w[HID + c];
  float ph = x0 * p1w[c] + x1 * p1w[HID + c];
  TH[i] = th;
  BT[i] = ph - th + t1b[c] + p1b[c];
}

// ---------------- fp32 WMMA GEMM (compile-time K,F): Out = X@W (+bias, relu) ----
// One wave: 64x32 output tile, 8 accumulators, each A fragment reused by two
// column tiles. Out-of-range columns CLAMPED (dup of col F-1, dropped by guarded
// store) -> no branches/selects in hot loop. Constant K/F put all load offsets
// into the 24-bit VMEM immediate field (no per-iteration 64-bit address math).
// rows must be a multiple of 256 (4 waves/block); K a multiple of 4.
template <int K, int F, int RELU>
__global__ __launch_bounds__(128) void k_gemm_wmma(
    const float* __restrict__ X, const float* __restrict__ W,
    const float* __restrict__ bias, float* __restrict__ Out) {
  const int lane = threadIdx.x & 31;
  const int wave = threadIdx.x >> 5;
  const int m    = lane & 15;
  const int kh   = lane >> 4;
  const int col0 = blockIdx.x * 32 + m;
  const int col1 = col0 + 16;
  const bool ok0 = col0 < F, ok1 = col1 < F;
  const int cc0 = ok0 ? col0 : (F - 1);
  const int cc1 = ok1 ? col1 : (F - 1);
  const size_t Rbase = (size_t)blockIdx.y * 256 + (size_t)wave * 64;

  const float* ap  = X + (Rbase + (size_t)m) * K + kh * 2;  // lane's K-pair base
  const float* w0p = W + kh * 2 * F + cc0;
  const float* w1p = W + kh * 2 * F + cc1;

  v8f c00 = {}, c01 = {}, c02 = {}, c03 = {};
  v8f c10 = {}, c11 = {}, c12 = {}, c13 = {};
#pragma unroll 4
  for (int k0 = 0; k0 < K; k0 += 4) {
    v2f b0, b1;
    b0.x = w0p[k0 * F]; b0.y = w0p[k0 * F + F];
    b1.x = w1p[k0 * F]; b1.y = w1p[k0 * F + F];
    v2f a0 = *(const v2f*)(ap + k0);
    v2f a1 = *(const v2f*)(ap + 16 * K + k0);
    v2f a2 = *(const v2f*)(ap + 32 * K + k0);
    v2f a3 = *(const v2f*)(ap + 48 * K + k0);
    c00 = __builtin_amdgcn_wmma_f32_16x16x4_f32(false, a0, false, b0, (short)0, c00, false, false);
    c01 = __builtin_amdgcn_wmma_f32_16x16x4_f32(false, a1, false, b0, (short)0, c01, false, false);
    c02 = __builtin_amdgcn_wmma_f32_16x16x4_f32(false, a2, false, b0, (short)0, c02, false, false);
    c03 = __builtin_amdgcn_wmma_f32_16x16x4_f32(false, a3, false, b0, (short)0, c03, false, false);
    c10 = __builtin_amdgcn_wmma_f32_16x16x4_f32(false, a0, false, b1, (short)0, c10, false, false);
    c11 = __builtin_amdgcn_wmma_f32_16x16x4_f32(false, a1, false, b1, (short)0, c11, false, false);
    c12 = __builtin_amdgcn_wmma_f32_16x16x4_f32(false, a2, false, b1, (short)0, c12, false, false);
    c13 = __builtin_amdgcn_wmma_f32_16x16x4_f32(false, a3, false, b1, (short)0, c13, false, false);
  }

  const float bv0 = (bias != nullptr) ? bias[cc0] : 0.0f;
  const float bv1 = (bias != nullptr) ? bias[cc1] : 0.0f;
  float* o0 = Out + (Rbase + (size_t)kh * 8) * F;
#pragma unroll
  for (int r = 0; r < 8; r++) {
    float v;
    if (ok0) {
      v = c00[r] + bv0; if (RELU) v = fmaxf(v, 0.0f); o0[(r     ) * F + col0] = v;
      v = c01[r] + bv0; if (RELU) v = fmaxf(v, 0.0f); o0[(r + 16) * F + col0] = v;
      v = c02[r] + bv0; if (RELU) v = fmaxf(v, 0.0f); o0[(r + 32) * F + col0] = v;
      v = c03[r] + bv0; if (RELU) v = fmaxf(v, 0.0f); o0[(r + 48) * F + col0] = v;
    }
    if (ok1) {
      v = c10[r] + bv1; if (RELU) v = fmaxf(v, 0.0f); o0[(r     ) * F + col1] = v;
      v = c11[r] + bv1; if (RELU) v = fmaxf(v, 0.0f); o0[(r + 16) * F + col1] = v;
      v = c12[r] + bv1; if (RELU) v = fmaxf(v, 0.0f); o0[(r + 32) * F + col1] = v;
      v = c13[r] + bv1; if (RELU) v = fmaxf(v, 0.0f); o0[(r + 48) * F + col1] = v;
    }
  }
}

// ---------------- per-node gather-max aggregation ----------------
__global__ __launch_bounds__(256) void k_agg(const float* __restrict__ TH,
                                             const float* __restrict__ BT,
                                             const int* __restrict__ rowptr,
                                             const int* __restrict__ csr,
                                             float* __restrict__ Hout, int N) {
  const int lane = threadIdx.x & 31;
  const int n = blockIdx.x * 8 + (threadIdx.x >> 5);
  if (n >= N) return;  // uniform per wave
  const int e0 = rowptr[n], e1 = rowptr[n + 1];
  const float4* THv = (const float4*)TH;
  const float NEG = -__builtin_inff();
  float4 mv; mv.x = NEG; mv.y = NEG; mv.z = NEG; mv.w = NEG;
  for (int e = e0; e < e1; ++e) {
    int s = csr[e];
    float4 v = THv[(size_t)s * 32 + lane];
    mv.x = fmaxf(mv.x, v.x); mv.y = fmaxf(mv.y, v.y);
    mv.z = fmaxf(mv.z, v.z); mv.w = fmaxf(mv.w, v.w);
  }
  float4 o;
  if (e1 > e0) {
    float4 b = ((const float4*)BT)[(size_t)n * 32 + lane];
    o.x = mv.x + b.x; o.y = mv.y + b.y; o.z = mv.z + b.z; o.w = mv.w + b.w;
  } else {
    o.x = 0.f; o.y = 0.f; o.z = 0.f; o.w = 0.f;
  }
  ((float4*)Hout)[(size_t)n * 32 + lane] = o;
}

// ---------------- final dot (200 -> 1) + per-graph mean ----------------
__global__ __launch_bounds__(256) void k_final(const float* __restrict__ M2,
                                               const float* __restrict__ ow,
                                               const float* __restrict__ ob,
                                               float* __restrict__ out,
                                               int baseNode, int rows, int Kf, int npg) {
  const int lane = threadIdx.x & 31;
  const int r = blockIdx.x * 8 + (threadIdx.x >> 5);
  if (r >= rows) return;  // uniform per wave
  const float* row = M2 + (size_t)r * Kf;
  float acc = 0.0f;
  for (int k = lane; k < Kf; k += 32) acc += row[k] * ow[k];
  for (int off = 16; off > 0; off >>= 1) acc += __shfl_xor(acc, off, 32);
  if (lane == 0) {
    int node = baseNode + r;
    atomicAdd(&out[node / npg], (acc + ob[0]) * (1.0f / (float)npg));
  }
}

// ---------------- host side ----------------
extern "C" void kernel_launch(void* const* d_in, const int* in_sizes, int n_in,
                              void* d_out, int out_size, void* d_ws, size_t ws_size,
                              hipStream_t stream) {
  const float* coords = (const float*)d_in[0];
  const float* t1w = (const float*)d_in[1];  const float* t1b = (const float*)d_in[2];
  const float* p1w = (const float*)d_in[3];  const float* p1b = (const float*)d_in[4];
  const float* t2w = (const float*)d_in[5];  const float* t2b = (const float*)d_in[6];
  const float* p2w = (const float*)d_in[7];  const float* p2b = (const float*)d_in[8];
  const float* t3w = (const float*)d_in[9];  const float* t3b = (const float*)d_in[10];
  const float* lw  = (const float*)d_in[13]; const float* lb  = (const float*)d_in[14];
  const float* l2w = (const float*)d_in[15]; const float* l2b = (const float*)d_in[16];
  const float* ow  = (const float*)d_in[17]; const float* ob  = (const float*)d_in[18];
  const float* p3w = (const float*)d_in[11]; const float* p3b = (const float*)d_in[12];
  const int* srcI  = (const int*)d_in[19];
  const int* dstI  = (const int*)d_in[20];
  (void)n_in; (void)ws_size;

  const int N = in_sizes[0] / 2;    // 51200
  const int E = in_sizes[19];       // 409600
  const int B = out_size;           // 64
  const int npg = N / B;            // 800 (graph_ids = arange // npg)

  char* wsb = (char*)d_ws; size_t off = 0;
  auto carve = [&](size_t bytes) -> void* {
    void* p = wsb + off;
    off = (off + bytes + 255) & ~(size_t)255;
    return p;
  };
  float* X  = (float*)carve((size_t)N * HID * 4);
  float* TH = (float*)carve((size_t)N * HID * 4);
  float* BT = (float*)carve((size_t)N * HID * 4);
  float* Wc = (float*)carve((size_t)HID * HID * 4);
  float* Bc = (float*)carve((size_t)HID * 4);
  const int CH = 6400;
  float* M1 = (float*)carve((size_t)CH * 400 * 4);
  float* M2 = (float*)carve((size_t)CH * 200 * 4);
  int* cnt    = (int*)carve((size_t)N * 4);
  int* rowptr = (int*)carve((size_t)(N + 1) * 4);
  int* cursor = (int*)carve((size_t)N * 4);
  int* csr    = (int*)carve((size_t)E * 4);

  // ---- CSR of incoming edges (built once, reused by all 3 layers) ----
  k_zero_i32<<<(N + 255) / 256, 256, 0, stream>>>(cnt, N);
  k_count<<<(E + 255) / 256, 256, 0, stream>>>(dstI, cnt, E);
  k_scan<<<1, 1024, 0, stream>>>(cnt, rowptr, N);
  k_copy_i32<<<(N + 255) / 256, 256, 0, stream>>>(rowptr, cursor, N);
  k_scatter<<<(E + 255) / 256, 256, 0, stream>>>(srcI, dstI, cursor, csr, E);

  // ---- layer 1 (K=2 scalar) + aggregation ----
  k_layer1<<<(N * HID + 255) / 256, 256, 0, stream>>>(coords, t1w, t1b, p1w, p1b, TH, BT, N * HID);
  k_agg<<<(N + 7) / 8, 256, 0, stream>>>(TH, BT, rowptr, csr, X, N);

  // ---- layers 2,3: two fp32-WMMA GEMMs + aggregation each ----
  const float* TWs[2] = {t2w, t3w}; const float* TBs[2] = {t2b, t3b};
  const float* PWs[2] = {p2w, p3w}; const float* PBs[2] = {p2b, p3b};
  for (int l = 0; l < 2; ++l) {
    k_wcombine<<<(HID * HID + 255) / 256, 256, 0, stream>>>(TWs[l], PWs[l], Wc, HID * HID);
    k_bcombine<<<1, 256, 0, stream>>>(TBs[l], PBs[l], Bc, HID);
    dim3 g((HID + 31) / 32, N / 256);
    k_gemm_wmma<HID, HID, 0><<<g, 128, 0, stream>>>(X, TWs[l], nullptr, TH);
    k_gemm_wmma<HID, HID, 0><<<g, 128, 0, stream>>>(X, Wc, Bc, BT);
    k_agg<<<(N + 7) / 8, 256, 0, stream>>>(TH, BT, rowptr, csr, X, N);
  }

  // ---- MLP 128->400->200->1 (chunked), then per-graph mean ----
  k_zero_f32<<<(B + 255) / 256, 256, 0, stream>>>((float*)d_out, B);
  for (int c0 = 0; c0 < N; c0 += CH) {
    k_gemm_wmma<HID, 400, 1><<<dim3((400 + 31) / 32, CH / 256), 128, 0, stream>>>(
        X + (size_t)c0 * HID, lw, lb, M1);
    k_gemm_wmma<400, 200, 1><<<dim3((200 + 31) / 32, CH / 256), 128, 0, stream>>>(
        M1, l2w, l2b, M2);
    k_final<<<CH / 8, 256, 0, stream>>>(M2, ow, ob, (float*)d_out, c0, CH, 200, npg);
  }
}